// ToBEVHeightCompression_8529805050239
// MI455X (gfx1250) — compile-verified
//
#include <hip/hip_runtime.h>

// Problem constants (from reference)
#define BEV0  512
#define HDIM  16
#define BEV1  512
#define BATCH 2
#define NPTS  300000
#define CF    16

// Output: (BATCH, HDIM*CF, BEV0, BEV1) = (2, 256, 512, 512) fp32
constexpr int CH        = HDIM * CF;                  // 256
constexpr int PLANE     = BEV0 * BEV1;                // 262144
constexpr int BATCH_STR = CH * PLANE;                 // 67108864
constexpr int OUT_ELEMS = BATCH * BATCH_STR;          // 134217728

typedef float __attribute__((ext_vector_type(4))) v4f;

// ---------------------------------------------------------------------------
// Kernel 1: streaming zero-fill of the 512 MB output with b128 stores.
// 8.4M threads x one v4f each, 4 strided iterations => fully coalesced.
// ---------------------------------------------------------------------------
__global__ void bev_zero_kernel(v4f* __restrict__ out, int n4) {
    const int stride = gridDim.x * blockDim.x;
    int i = blockIdx.x * blockDim.x + threadIdx.x;
    const v4f z = {0.0f, 0.0f, 0.0f, 0.0f};
#pragma unroll 4
    for (int k = 0; k < 4; ++k) {
        if (i < n4) out[i] = z;   // global_store_b128
        i += stride;
    }
}

// ---------------------------------------------------------------------------
// Kernel 2: scatter-add. Thread t -> point p = t>>4, feature f = t&15.
//  - feats read is 128B-contiguous per wave32 (coalesced)
//  - coords read is one int4 (b128) broadcast across 16 lanes
//  - accumulation uses the native CDNA5 no-return fp32 global atomic
//    (L2-side RMW, tracked on STOREcnt; s_endpgm's implicit wait covers it)
// ---------------------------------------------------------------------------
__global__ void bev_scatter_kernel(const int4*  __restrict__ coords,
                                   const float* __restrict__ feats,
                                   float*       __restrict__ out) {
    const int t = blockIdx.x * blockDim.x + threadIdx.x;
    if (t >= NPTS * CF) return;

    const int p = t >> 4;     // point index
    const int f = t & 15;     // feature index

    const int4 c = coords[p];             // x = c.x, y = c.y, z = c.z, b = c.w
    int y = c.y;
    y = (y < 0) ? 0 : ((y > HDIM - 1) ? (HDIM - 1) : y);

    const int  ch  = y * CF + f;                              // channel
    const int  idx = c.w * BATCH_STR + ch * PLANE + c.x * BEV1 + c.z;
    const float v  = feats[t];                                // == feats[p*CF + f]

    float* addr = out + idx;
    // Native fp32 atomic add, no return, device scope (RMW at L2).
    asm volatile("global_atomic_add_f32 %0, %1, off scope:SCOPE_DEV"
                 :: "v"(addr), "v"(v)
                 : "memory");
}

// ---------------------------------------------------------------------------
extern "C" void kernel_launch(void* const* d_in, const int* in_sizes, int n_in,
                              void* d_out, int out_size, void* d_ws, size_t ws_size,
                              hipStream_t stream) {
    const int4*  coords = (const int4*)d_in[0];   // (NPTS, 4) int32
    const float* feats  = (const float*)d_in[1];  // (NPTS, 16) fp32
    float*       out    = (float*)d_out;          // (2, 256, 512, 512) fp32

    // 1) zero the output (poisoned by harness; must be re-zeroed every call)
    const int n4      = OUT_ELEMS / 4;            // 33,554,432 float4s
    const int zblocks = n4 / (256 * 4);           // 32768 blocks, exact cover
    bev_zero_kernel<<<zblocks, 256, 0, stream>>>((v4f*)out, n4);

    // 2) scatter-accumulate features
    const int total  = NPTS * CF;                 // 4,800,000 threads
    const int blocks = (total + 255) / 256;
    bev_scatter_kernel<<<blocks, 256, 0, stream>>>(coords, feats, out);
}